// TwistorResonance_72241349919091
// MI455X (gfx1250) — compile-verified
//
#include <hip/hip_runtime.h>
#include <math.h>

// TwistorResonance for MI455X (gfx1250, wave32).
// cos((phi_i-phi_j)*s) = c_i*c_j + s_i*s_j  -> rank-2 GEMM -> V_WMMA_F32_16X16X4_F32
// Top-k threshold via 3-level radix select on float bits (2048-bin LDS histograms).
// All reductions integer-atomic or fixed-order -> deterministic.

typedef float v2f __attribute__((ext_vector_type(2)));
typedef float v8f __attribute__((ext_vector_type(8)));

#define NN   4096
#define BB   2
#define RANK0 32769u   /* SPARSE_K*N + 1 : rank of threshold element (descending) */

__device__ __forceinline__ float softplusf(float x) { return log1pf(__expf(x)); }

__device__ __forceinline__ v8f wmma_cosdphi(v2f a, v2f b) {
  v8f c = {};
  // D = A(16x4) * B(4x16) + 0 ; K=2..3 zero-padded via lane mask
  return __builtin_amdgcn_wmma_f32_16x16x4_f32(false, a, false, b, (short)0, c, false, false);
}

// ---------------------------------------------------------------------------
// K0: amp / interleaved (cos,sin)(s*phi), zero hist + ctrl + out
// ---------------------------------------------------------------------------
__global__ __launch_bounds__(256) void tw_prep(const float* __restrict__ zr,
                                               const float* __restrict__ zi,
                                               const float* __restrict__ ps,
                                               float* __restrict__ amp,
                                               float* __restrict__ csv,
                                               unsigned* __restrict__ hist,
                                               unsigned* __restrict__ ctrl,
                                               float* __restrict__ out) {
  int idx = blockIdx.x * 256 + threadIdx.x;           // 0 .. 16383
  float sens = ps[0];
  if (idx < BB * NN) {
    float x = zr[idx], y = zi[idx];
    amp[idx] = sqrtf(x * x + y * y);
    float phi = atan2f(y, x) * sens;
    csv[2 * idx + 0] = cosf(phi);
    csv[2 * idx + 1] = sinf(phi);
  }
  if (idx < BB * 2048) hist[idx] = 0u;
  if (idx < BB) {
    ctrl[idx * 4 + 0] = 0u;      // prefix (becomes thr bits after level 2)
    ctrl[idx * 4 + 1] = RANK0;   // remaining rank
    ctrl[idx * 4 + 2] = 0u;
    ctrl[idx * 4 + 3] = 0u;
  }
  if (idx < BB * NN * 2) out[idx] = 0.0f;
}

// ---------------------------------------------------------------------------
// K1: histogram pass over R bits (one radix level). WMMA tile = 16x16 cos-dphi.
// grid: (N/16, B), block: 256 = 8 waves; wave w does j-tiles w, w+8, ...
// Templated on LEVEL: level-0 pass has NO per-element candidate branch and
// all shifts/masks are immediates.
// ---------------------------------------------------------------------------
template <int LEVEL>
__global__ __launch_bounds__(256) void tw_hist(const float* __restrict__ csv,
                                               const float* __restrict__ amp,
                                               const float* __restrict__ ksp,
                                               const float* __restrict__ kbp,
                                               unsigned* __restrict__ hist,
                                               const unsigned* __restrict__ ctrl) {
  constexpr int      SHIFT  = (LEVEL == 0) ? 21 : (LEVEL == 1) ? 10 : 0;
  constexpr unsigned BMASK  = (LEVEL == 2) ? 0x3FFu : 0x7FFu;
  constexpr int      CSHIFT = (LEVEL == 1) ? 21 : 10;   // unused for LEVEL==0

  __shared__ float lCS[2 * NN];
  __shared__ float lA[NN];
  __shared__ unsigned lh[2048];

  const int i0    = blockIdx.x * 16;
  const int batch = blockIdx.y;
  const int tid   = threadIdx.x;
  const int lane  = tid & 31;
  const int wv    = tid >> 5;
  const int ln15  = lane & 15;
  const float hmask = (lane >= 16) ? 0.0f : 1.0f;      // zero K=2..3 rows of A/B
  const bool hi   = lane >= 16;

  const float* gCS = csv + batch * 2 * NN;
  const float* gA  = amp + batch * NN;
  for (int t = tid; t < 2 * NN; t += 256) lCS[t] = gCS[t];
  for (int t = tid; t < NN; t += 256) lA[t] = gA[t];
  for (int t = tid; t < 2048; t += 256) lh[t] = 0u;
  __syncthreads();

  const float e = softplusf(ksp[0]) + softplusf(kbp[0]) + 1e-6f;
  const unsigned prefHi = ctrl[batch * 4 + 0] >> CSHIFT;

  v2f afrag = *(const v2f*)&lCS[2 * (i0 + ln15)];
  afrag.x *= hmask;
  afrag.y *= hmask;
  float ai[8];
  const int ib = i0 + (hi ? 8 : 0);
#pragma unroll
  for (int k = 0; k < 8; ++k) ai[k] = lA[ib + k];

  for (int jt = wv; jt < NN / 16; jt += 8) {
    const int j0 = jt * 16;
    v2f bf = *(const v2f*)&lCS[2 * (j0 + ln15)];
    bf.x *= hmask;
    bf.y *= hmask;
    const float aj = lA[j0 + ln15];
    v8f d = wmma_cosdphi(afrag, bf);
#pragma unroll
    for (int k = 0; k < 8; ++k) {
      float t = fmaxf(fabsf(d[k]), 1e-10f);
      float p = __builtin_amdgcn_exp2f(e * __builtin_amdgcn_logf(t));
      float r = ai[k] * aj * p;
      unsigned bits = __float_as_uint(r);
      if (LEVEL == 0) {
        atomicAdd(&lh[(bits >> SHIFT) & BMASK], 1u);
      } else {
        if ((bits >> CSHIFT) == prefHi) atomicAdd(&lh[(bits >> SHIFT) & BMASK], 1u);
      }
    }
  }
  __syncthreads();
  for (int t = tid; t < 2048; t += 256) {
    unsigned v = lh[t];
    if (v) atomicAdd(&hist[batch * 2048 + t], v);
  }
}

// ---------------------------------------------------------------------------
// K2: scan histogram descending, pick bin crossing remaining rank, update
// prefix/rank, zero hist for next level.  grid: (B), block: 256
// ---------------------------------------------------------------------------
__global__ __launch_bounds__(256) void tw_scan(unsigned* __restrict__ hist,
                                               unsigned* __restrict__ ctrl,
                                               int level, int shift) {
  const int b = blockIdx.x;
  unsigned* h = hist + b * 2048;
  unsigned* c = ctrl + b * 4;
  const int nbins = (level == 2) ? 1024 : 2048;

  __shared__ unsigned csum[256];
  const int t = threadIdx.x;
  const int base = t * 8;
  unsigned s = 0u;
  if (base < nbins) {
#pragma unroll
    for (int q = 0; q < 8; ++q) s += h[base + q];
  }
  csum[t] = s;
  __syncthreads();

  if (t == 0) {
    unsigned rank = c[1];
    unsigned cum = 0u;
    unsigned bin = 0u;
    for (int ch = nbins / 8 - 1; ch >= 0; --ch) {
      if (cum + csum[ch] >= rank) {
        for (int bi = ch * 8 + 7; bi >= ch * 8; --bi) {
          unsigned hv = h[bi];
          if (cum + hv >= rank) { bin = (unsigned)bi; break; }
          cum += hv;
        }
        break;
      }
      cum += csum[ch];
    }
    c[0] = c[0] | (bin << (shift & 31));
    c[1] = rank - cum;
  }
  __syncthreads();
  if (base < nbins) {
#pragma unroll
    for (int q = 0; q < 8; ++q) h[base + q] = 0u;
  }
}

// ---------------------------------------------------------------------------
// K3: recompute R tiles (WMMA), threshold, accumulate out = alpha * R_sp @ z.
// Each block owns one 16-row tile -> writes its rows exclusively (no atomics).
// ---------------------------------------------------------------------------
__global__ __launch_bounds__(256) void tw_final(const float* __restrict__ csv,
                                                const float* __restrict__ amp,
                                                const float* __restrict__ zr,
                                                const float* __restrict__ zi,
                                                const float* __restrict__ ksp,
                                                const float* __restrict__ kbp,
                                                const float* __restrict__ rsp,
                                                const unsigned* __restrict__ ctrl,
                                                float* __restrict__ out) {
  __shared__ float lCS[2 * NN];
  __shared__ float lA[NN];
  __shared__ float partial[8][16][2];

  const int i0    = blockIdx.x * 16;
  const int batch = blockIdx.y;
  const int tid   = threadIdx.x;
  const int lane  = tid & 31;
  const int wv    = tid >> 5;
  const int ln15  = lane & 15;
  const float hmask = (lane >= 16) ? 0.0f : 1.0f;
  const bool hi   = lane >= 16;

  const float* gCS = csv + batch * 2 * NN;
  const float* gA  = amp + batch * NN;
  const float* gZr = zr + batch * NN;
  const float* gZi = zi + batch * NN;
  for (int t = tid; t < 2 * NN; t += 256) lCS[t] = gCS[t];
  for (int t = tid; t < NN; t += 256) lA[t] = gA[t];
  __syncthreads();

  const float e     = softplusf(ksp[0]) + softplusf(kbp[0]) + 1e-6f;
  const float alpha = rsp[0];
  const float thr   = __uint_as_float(ctrl[batch * 4 + 0]);

  v2f afrag = *(const v2f*)&lCS[2 * (i0 + ln15)];
  afrag.x *= hmask;
  afrag.y *= hmask;
  float ai[8];
  const int ib = i0 + (hi ? 8 : 0);
#pragma unroll
  for (int k = 0; k < 8; ++k) ai[k] = lA[ib + k];

  float accR[8], accI[8];
#pragma unroll
  for (int k = 0; k < 8; ++k) { accR[k] = 0.0f; accI[k] = 0.0f; }

  for (int jt = wv; jt < NN / 16; jt += 8) {
    const int j0 = jt * 16;
    v2f bf = *(const v2f*)&lCS[2 * (j0 + ln15)];
    bf.x *= hmask;
    bf.y *= hmask;
    const float aj  = lA[j0 + ln15];
    const float zrj = gZr[j0 + ln15];
    const float zij = gZi[j0 + ln15];
    v8f d = wmma_cosdphi(afrag, bf);
#pragma unroll
    for (int k = 0; k < 8; ++k) {
      float t = fmaxf(fabsf(d[k]), 1e-10f);
      float p = __builtin_amdgcn_exp2f(e * __builtin_amdgcn_logf(t));
      float r = ai[k] * aj * p;
      r = (r >= thr) ? r : 0.0f;
      accR[k] = fmaf(r, zrj, accR[k]);
      accI[k] = fmaf(r, zij, accI[k]);
    }
  }

  // Reduce across the 16 lanes of each half (cols) -> rows k (low), k+8 (high).
#pragma unroll
  for (int k = 0; k < 8; ++k) {
    float vr = accR[k], vi = accI[k];
#pragma unroll
    for (int m = 1; m < 16; m <<= 1) {
      vr += __shfl_xor(vr, m, 32);
      vi += __shfl_xor(vi, m, 32);
    }
    if (ln15 == 0) {
      int row = k + (hi ? 8 : 0);
      partial[wv][row][0] = vr;
      partial[wv][row][1] = vi;
    }
  }
  __syncthreads();

  // Deterministic fixed-order cross-wave reduction + store.
  if (tid < 32) {
    int row = tid >> 1, comp = tid & 1;
    float sum = 0.0f;
    for (int w8 = 0; w8 < 8; ++w8) sum += partial[w8][row][comp];
    out[((size_t)batch * NN + (size_t)(i0 + row)) * 2 + comp] = alpha * sum;
  }
}

// ---------------------------------------------------------------------------
extern "C" void kernel_launch(void* const* d_in, const int* in_sizes, int n_in,
                              void* d_out, int out_size, void* d_ws, size_t ws_size,
                              hipStream_t stream) {
  (void)in_sizes; (void)n_in; (void)out_size; (void)ws_size;
  const float* zr = (const float*)d_in[0];
  const float* zi = (const float*)d_in[1];
  const float* rs = (const float*)d_in[2];
  const float* ks = (const float*)d_in[3];
  const float* kb = (const float*)d_in[4];
  const float* ps = (const float*)d_in[5];
  float* out = (float*)d_out;

  float* w = (float*)d_ws;
  float*    amp  = w;                               // B*N
  float*    csv  = w + BB * NN;                     // 2*B*N interleaved (cos,sin)
  unsigned* hist = (unsigned*)(w + 3 * BB * NN);    // B*2048
  unsigned* ctrl = hist + BB * 2048;                // B*4

  tw_prep<<<dim3(64), dim3(256), 0, stream>>>(zr, zi, ps, amp, csv, hist, ctrl, out);

  // Radix-select levels: bits [31:21], [20:10], [9:0]
  tw_hist<0><<<dim3(NN / 16, BB), dim3(256), 0, stream>>>(csv, amp, ks, kb, hist, ctrl);
  tw_scan<<<dim3(BB), dim3(256), 0, stream>>>(hist, ctrl, 0, 21);
  tw_hist<1><<<dim3(NN / 16, BB), dim3(256), 0, stream>>>(csv, amp, ks, kb, hist, ctrl);
  tw_scan<<<dim3(BB), dim3(256), 0, stream>>>(hist, ctrl, 1, 10);
  tw_hist<2><<<dim3(NN / 16, BB), dim3(256), 0, stream>>>(csv, amp, ks, kb, hist, ctrl);
  tw_scan<<<dim3(BB), dim3(256), 0, stream>>>(hist, ctrl, 2, 0);

  tw_final<<<dim3(NN / 16, BB), dim3(256), 0, stream>>>(
      csv, amp, zr, zi, ks, kb, rs, ctrl, out);
}